// GCNLayer_38431367365104
// MI455X (gfx1250) — compile-verified
//
#include <hip/hip_runtime.h>

typedef __attribute__((ext_vector_type(2))) float v2f;
typedef __attribute__((ext_vector_type(8))) float v8f;

#define B_   4
#define N_   4096
#define D_   128
#define R_   4
#define K_   16
#define O_   128
#define PAD_ 132          // padded LDS row stride (floats) -> conflict-free WMMA A reads
#define NMAT 5            // 4 relation matrices + 1 self matrix

__global__ __launch_bounds__(256)
void gcn_rel_wmma(const float* __restrict__ feat,
                  const int*   __restrict__ nidx,
                  const float* __restrict__ relW,
                  const float* __restrict__ selfW,
                  const float* __restrict__ bias,
                  float*       __restrict__ out)
{
    __shared__ __align__(16) float ldsA[NMAT * 16 * PAD_];   // ~42 KB

    const int tile = blockIdx.x;        // B*N/16 tiles
    const int b    = tile >> 8;         // 256 node-tiles per batch
    const int n0   = (tile & 255) << 4;

    const int lane = threadIdx.x & 31;
    const int wave = threadIdx.x >> 5;

    const float* featB = feat + (size_t)b * N_ * D_;

    // -------- Phase 1: gather + mean (relations) and self rows -> LDS --------
    // One task = one 128-float row; 32 lanes x float4 = full row, coalesced 512B.
    for (int task = wave; task < NMAT * 16; task += 8) {
        const int mat = task >> 4;      // 0..3 = relation r, 4 = self
        const int m   = task & 15;
        float* dst = &ldsA[(mat * 16 + m) * PAD_ + lane * 4];
        if (mat < 4) {
            const int* idxp = nidx + ((size_t)(b * R_ + mat) * N_ + (n0 + m)) * K_;
            float ax = 0.f, ay = 0.f, az = 0.f, aw = 0.f;
            #pragma unroll
            for (int kq = 0; kq < 4; ++kq) {
                const int4 iv = ((const int4*)idxp)[kq];
                const int ids[4] = {iv.x, iv.y, iv.z, iv.w};
                #pragma unroll
                for (int j = 0; j < 4; ++j) {
                    const int idx = ids[j];
                    if (idx > 0) {      // idx 0 = zero-pad row
                        const float4 v =
                            *(const float4*)(featB + (size_t)(idx - 1) * D_ + lane * 4);
                        ax += v.x; ay += v.y; az += v.z; aw += v.w;
                    }
                }
            }
            const float s = 1.0f / (float)K_;
            dst[0] = ax * s; dst[1] = ay * s; dst[2] = az * s; dst[3] = aw * s;
        } else {
            const float4 v = *(const float4*)(featB + (size_t)(n0 + m) * D_ + lane * 4);
            dst[0] = v.x; dst[1] = v.y; dst[2] = v.z; dst[3] = v.w;
        }
    }

    __syncthreads();

    // -------- Phase 2: D += A(16xD) x W(DxO) via V_WMMA_F32_16X16X4_F32 ------
    // Wave w owns output columns [16w, 16w+16). ISA 16x4 f32 A layout:
    //   lanes 0-15: M=lane, K = d..d+1 ; lanes 16-31: M=lane-16, K = d+2..d+3
    const int col   = lane & 15;
    const int koff  = (lane >> 4) * 2;
    const int obase = wave * 16 + col;

    v8f acc;
    {
        const float bv = bias[obase];   // bias replicated down the M rows
        #pragma unroll
        for (int j = 0; j < 8; ++j) acc[j] = bv;
    }

    #pragma unroll
    for (int mat = 0; mat < NMAT; ++mat) {
        const float* W  = (mat < 4) ? (relW + (size_t)mat * D_ * O_) : selfW;
        const float* Ap = &ldsA[(mat * 16 + col) * PAD_ + koff];
        const float* Wp = W + (size_t)koff * O_ + obase;
        #pragma unroll 8
        for (int kk = 0; kk < D_ / 4; ++kk) {
            const int d = kk * 4;
            v2f a, bb;
            a.x  = Ap[d];                       // ds_load_b64, conflict-free
            a.y  = Ap[d + 1];
            bb.x = Wp[(size_t)d * O_];          // weights are L2/L0 resident
            bb.y = Wp[(size_t)(d + 1) * O_];
            acc = __builtin_amdgcn_wmma_f32_16x16x4_f32(
                false, a, false, bb, (short)0, acc, false, false);
        }
    }

    // -------- ReLU + store (C/D layout: VGPR j -> rows j and j+8) ------------
    float* outp = out + ((size_t)b * N_ + n0) * O_ + obase;
    const int rhalf = (lane >> 4) * 8;
    #pragma unroll
    for (int j = 0; j < 8; ++j) {
        float v = acc[j];
        v = v > 0.f ? v : 0.f;
        outp[(size_t)(j + rhalf) * O_] = v;
    }
}

extern "C" void kernel_launch(void* const* d_in, const int* in_sizes, int n_in,
                              void* d_out, int out_size, void* d_ws, size_t ws_size,
                              hipStream_t stream) {
    const float* feat  = (const float*)d_in[0];   // [B,N,D] f32
    const int*   nidx  = (const int*)d_in[1];     // [B,R,N,K] i32
    const float* relW  = (const float*)d_in[2];   // [R,D,O] f32
    const float* selfW = (const float*)d_in[3];   // [D,O] f32
    const float* bias  = (const float*)d_in[4];   // [O] f32
    float* outp = (float*)d_out;                  // [B,N,O] f32
    (void)in_sizes; (void)n_in; (void)out_size; (void)d_ws; (void)ws_size;

    dim3 grid(B_ * N_ / 16);   // 1024 workgroups, one 16-node tile each
    dim3 block(256);           // 8 wave32; wave w -> O columns [16w,16w+16)
    gcn_rel_wmma<<<grid, block, 0, stream>>>(feat, nidx, relW, selfW, bias, outp);
}